// HashEmbedder_68994354643585
// MI455X (gfx1250) — compile-verified
//
#include <hip/hip_runtime.h>
#include <stdint.h>

#define N_POINTS  1048576
#define N_LEVELS  16
#define LOG2_T    19
#define TBL_MASK  ((1u << LOG2_T) - 1u)
#define PRIME_Y   2654435761u
#define PRIME_Z   805459861u

// res_l = floor(16 * 2^(l/3))  (b = 32^(1/15) = cbrt(2))
__constant__ float c_res[N_LEVELS] = {
    16.f, 20.f, 25.f, 32.f, 40.f, 50.f, 64.f, 80.f,
    101.f, 128.f, 161.f, 203.f, 256.f, 322.f, 406.f, 512.f
};

__device__ __forceinline__ float2 lerp2(float2 a, float2 b, float w) {
    float2 r;
    r.x = fmaf(w, b.x - a.x, a.x);
    r.y = fmaf(w, b.y - a.y, a.y);
    return r;
}

// One lane = one (point, level) pair. 16 lanes per point -> wave32 covers
// 2 points x 16 levels; per-half-wave stores are one dense 128B segment.
// Block = 256 threads = 16 points.
__global__ __launch_bounds__(256) void hashgrid_kernel(
    const float* __restrict__ x,      // (N_POINTS, 3)
    const float* __restrict__ emb,    // (16, 2^19, 2) -- 64 MB, L2-resident
    const float* __restrict__ bbox,   // (2, 3)
    float* __restrict__ out,          // (N_POINTS, 32)
    float* __restrict__ mask)         // (N_POINTS,)
{
    __shared__ float sx[48];          // 16 points x 3 coords

    const unsigned t        = threadIdx.x;
    const unsigned blockPt0 = blockIdx.x * 16u;

    // ---- CDNA5 async global->LDS staging of this block's coordinates ----
    if (t < 48u) {
        uint64_t gaddr  = (uint64_t)(x + (size_t)blockPt0 * 3u + t);
        unsigned ldsOff = (unsigned)(uintptr_t)(&sx[t]);
        asm volatile("global_load_async_to_lds_b32 %0, %1, off"
                     :: "v"(ldsOff), "v"(gaddr) : "memory");
    }
    asm volatile("s_wait_asynccnt 0x0" ::: "memory");
    __syncthreads();

    const unsigned lp    = t >> 4;    // local point 0..15
    const unsigned level = t & 15u;   // 0..15
    const unsigned point = blockPt0 + lp;

    const float px = sx[lp * 3 + 0];
    const float py = sx[lp * 3 + 1];
    const float pz = sx[lp * 3 + 2];

    // bounding box (uniform across all lanes; scalarized to SMEM loads)
    const float bminx = bbox[0], bminy = bbox[1], bminz = bbox[2];
    const float bmaxx = bbox[3], bmaxy = bbox[4], bmaxz = bbox[5];

    if (level == 0u) {
        bool keep = (px >= bminx) & (px <= bmaxx) &
                    (py >= bminy) & (py <= bmaxy) &
                    (pz >= bminz) & (pz <= bmaxz);
        // write-once stream: non-temporal store, don't pollute L2
        __builtin_nontemporal_store(keep ? 1.0f : 0.0f, &mask[point]);
    }

    // clamp to box
    const float cx = fminf(fmaxf(px, bminx), bmaxx);
    const float cy = fminf(fmaxf(py, bminy), bmaxy);
    const float cz = fminf(fmaxf(pz, bminz), bmaxz);

    const float res = c_res[level];
    const float igx = res / (bmaxx - bminx);
    const float igy = res / (bmaxy - bminy);
    const float igz = res / (bmaxz - bminz);

    const float fx = (cx - bminx) * igx;
    const float fy = (cy - bminy) * igy;
    const float fz = (cz - bminz) * igz;

    const int bx = (int)floorf(fx);
    const int by = (int)floorf(fy);
    const int bz = (int)floorf(fz);

    // interpolation weights use the ORIGINAL (unclamped) coords, per reference
    const float wx = (px - bminx) * igx - (float)bx;
    const float wy = (py - bminy) * igy - (float)by;
    const float wz = (pz - bminz) * igz - (float)bz;

    // spatial hash of the 8 corners
    const uint32_t hx0 = (uint32_t)bx,            hx1 = (uint32_t)bx + 1u;
    const uint32_t hy0 = (uint32_t)by * PRIME_Y,  hy1 = ((uint32_t)by + 1u) * PRIME_Y;
    const uint32_t hz0 = (uint32_t)bz * PRIME_Z,  hz1 = ((uint32_t)bz + 1u) * PRIME_Z;

    const uint32_t h000 = (hx0 ^ hy0 ^ hz0) & TBL_MASK;  // v0 (0,0,0)
    const uint32_t h001 = (hx0 ^ hy0 ^ hz1) & TBL_MASK;  // v1 (0,0,1)
    const uint32_t h010 = (hx0 ^ hy1 ^ hz0) & TBL_MASK;  // v2 (0,1,0)
    const uint32_t h011 = (hx0 ^ hy1 ^ hz1) & TBL_MASK;  // v3 (0,1,1)
    const uint32_t h100 = (hx1 ^ hy0 ^ hz0) & TBL_MASK;  // v4 (1,0,0)
    const uint32_t h101 = (hx1 ^ hy0 ^ hz1) & TBL_MASK;  // v5 (1,0,1)
    const uint32_t h110 = (hx1 ^ hy1 ^ hz0) & TBL_MASK;  // v6 (1,1,0)
    const uint32_t h111 = (hx1 ^ hy1 ^ hz1) & TBL_MASK;  // v7 (1,1,1)

    // 8 independent b64 gathers — default RT policy keeps table in L2
    const float2* __restrict__ tbl =
        (const float2*)emb + ((size_t)level << LOG2_T);
    const float2 v0 = tbl[h000];
    const float2 v1 = tbl[h001];
    const float2 v2 = tbl[h010];
    const float2 v3 = tbl[h011];
    const float2 v4 = tbl[h100];
    const float2 v5 = tbl[h101];
    const float2 v6 = tbl[h110];
    const float2 v7 = tbl[h111];

    // trilinear blend
    const float2 c00 = lerp2(v0, v4, wx);
    const float2 c01 = lerp2(v1, v5, wx);
    const float2 c10 = lerp2(v2, v6, wx);
    const float2 c11 = lerp2(v3, v7, wx);
    const float2 c0  = lerp2(c00, c10, wy);
    const float2 c1  = lerp2(c01, c11, wy);
    const float2 r   = lerp2(c0, c1, wz);

    // coalesced dense 128B segment per half-wave; non-temporal (write-once
    // stream, 132 MB total) so it never evicts the L2-resident hash table
    union { float2 f2; unsigned long long u64; } cvt;
    cvt.f2 = r;
    __builtin_nontemporal_store(
        cvt.u64,
        (unsigned long long*)(out + (size_t)point * 32u + (size_t)level * 2u));
}

extern "C" void kernel_launch(void* const* d_in, const int* in_sizes, int n_in,
                              void* d_out, int out_size, void* d_ws, size_t ws_size,
                              hipStream_t stream) {
    (void)in_sizes; (void)n_in; (void)out_size; (void)d_ws; (void)ws_size;
    const float* x    = (const float*)d_in[0];
    const float* emb  = (const float*)d_in[1];
    const float* bbox = (const float*)d_in[2];

    float* out  = (float*)d_out;
    float* mask = out + (size_t)N_POINTS * 32u;   // tuple outputs concatenated

    const unsigned pairs  = N_POINTS * N_LEVELS;  // 16M lanes
    const unsigned blocks = pairs / 256u;         // 65536 blocks of 256
    hipLaunchKernelGGL(hashgrid_kernel, dim3(blocks), dim3(256), 0, stream,
                       x, emb, bbox, out, mask);
}